// BiLSTM_5970004542177
// MI455X (gfx1250) — compile-verified
//
#include <hip/hip_runtime.h>
#include <hip/hip_bf16.h>

// ---------------------------------------------------------------------------
// BiLSTM (2 layers, residual on layer 1, merge='ave') for MI455X / gfx1250.
// Strategy:
//   * Batch-sliced persistent scan: each workgroup owns 16 batch rows and one
//     direction; the full T=256 recurrence runs inside the workgroup using
//     only __syncthreads(). No cross-workgroup sync needed (recurrence is
//     independent across batch).
//   * Per step: z(16x1024) = [x_t ; h](16x512) @ [W;R](512x1024), computed
//     with V_WMMA_F32_16X16X4_F32 (full fp32 precision matrix pipe).
//   * Weights pre-packed into WMMA B-operand lane order -> inner loop does
//     one contiguous b64 global load per tile per K-step (L2 resident).
// ---------------------------------------------------------------------------

typedef float v2f __attribute__((ext_vector_type(2)));
typedef float v8f __attribute__((ext_vector_type(8)));

#define T_STEPS 256
#define UNITS_  256
#define N4      1024          // 4*U gate width
#define KTOT    512           // D + U
#define ASTRIDE 516           // padded LDS stride for [x;h] (bank spread)
#define ZSTRIDE 1028          // padded LDS stride for z     (bank spread)
#define COMBO_F (512 * 1024)  // floats per packed (layer,dir) weight block

__device__ __forceinline__ float sigmoid_f(float x) {
    return 1.0f / (1.0f + __expf(-x));
}
__device__ __forceinline__ float tanh_f(float x) {
    x = fminf(15.0f, fmaxf(-15.0f, x));
    float e = __expf(2.0f * x);
    return (e - 1.0f) / (e + 1.0f);
}

// ---------------------------------------------------------------------------
// Pack [W;R] (512 x 1024) into WMMA-B lane order:
//   packed[combo][nt(64)][ks(128)][lane(32)][2]
//   lane<16 : {M[4ks+0][16nt+lane],    M[4ks+1][16nt+lane]}
//   lane>=16: {M[4ks+2][16nt+lane-16], M[4ks+3][16nt+lane-16]}
// combo = layer*2 + (bw?1:0)
// ---------------------------------------------------------------------------
__global__ void pack_weights_kernel(const float* __restrict__ kfw,
                                    const float* __restrict__ rfw,
                                    const float* __restrict__ kbw,
                                    const float* __restrict__ rbw,
                                    float* __restrict__ packed) {
    int i = blockIdx.x * blockDim.x + threadIdx.x;  // 0 .. 4*64*128*32-1
    int lane  = i & 31;
    int ks    = (i >> 5) & 127;
    int nt    = (i >> 12) & 63;
    int combo = i >> 18;
    int layer = combo >> 1;
    int isbw  = combo & 1;
    const float* W = (isbw ? kbw : kfw) + (size_t)layer * 256 * 1024;
    const float* R = (isbw ? rbw : rfw) + (size_t)layer * 256 * 1024;

    int n  = nt * 16 + (lane & 15);
    int k0 = ks * 4 + ((lane >> 4) << 1);
    float d0, d1;
    if (k0 < 256) {
        d0 = W[(size_t)k0 * 1024 + n];
        d1 = W[(size_t)(k0 + 1) * 1024 + n];
    } else {
        int kk = k0 - 256;
        d0 = R[(size_t)kk * 1024 + n];
        d1 = R[(size_t)(kk + 1) * 1024 + n];
    }
    packed[(size_t)i * 2 + 0] = d0;
    packed[(size_t)i * 2 + 1] = d1;
}

// ---------------------------------------------------------------------------
// One bidirectional LSTM layer. grid = (8, 2): x = batch slice, y = direction.
// 1024 threads = 32 waves; wave w computes z columns [32w, 32w+32).
// ---------------------------------------------------------------------------
__global__ void __launch_bounds__(1024)
bilstm_layer_kernel(const float* __restrict__ in_fw,
                    const float* __restrict__ in_bw,
                    const float* __restrict__ pw_fw,
                    const float* __restrict__ pw_bw,
                    const float* __restrict__ bias_fw,
                    const float* __restrict__ bias_bw,
                    float* __restrict__ out_fw,
                    float* __restrict__ out_bw,
                    int residual) {
    extern __shared__ float lds[];
    float* Alds = lds;                 // 16 x ASTRIDE : cols [0,256)=x_t, [256,512)=h
    float* Zlds = lds + 16 * ASTRIDE;  // 16 x ZSTRIDE : gate pre-activations

    const int dir        = blockIdx.y;          // 0=fw, 1=bw
    const float* in      = dir ? in_bw   : in_fw;
    const float* pw      = dir ? pw_bw   : pw_fw;
    const float* bias    = dir ? bias_bw : bias_fw;
    float*       out     = dir ? out_bw  : out_fw;

    const int b0      = blockIdx.x * 16;
    const int tid     = threadIdx.x;
    const int lane    = tid & 31;
    const int wave    = tid >> 5;
    const int halfsel = lane >> 4;   // 0: lanes 0-15, 1: lanes 16-31
    const int lrow    = lane & 15;

    // --- per-thread elementwise ownership: 4 (row, unit) pairs -------------
    int   rr[4], uu[4];
    float bI[4], bF[4], bG[4], bO[4], cst[4];
#pragma unroll
    for (int j = 0; j < 4; ++j) {
        int f = tid + j * 1024;            // 0..4095
        rr[j] = f >> 8;
        uu[j] = f & 255;
        bI[j] = bias[uu[j]];
        bF[j] = bias[256 + uu[j]];
        bG[j] = bias[512 + uu[j]];
        bO[j] = bias[768 + uu[j]];
        cst[j] = 0.0f;
        Alds[rr[j] * ASTRIDE + 256 + uu[j]] = 0.0f;   // h_{-1} = 0
    }

    // --- GEMM constants ----------------------------------------------------
    const int nt0 = wave * 2;
    const int nt1 = wave * 2 + 1;
    const float* pwt0 = pw + ((size_t)(nt0 * 128) * 32 + lane) * 2;
    const float* pwt1 = pw + ((size_t)(nt1 * 128) * 32 + lane) * 2;
    const int aBase = lrow * ASTRIDE + 2 * halfsel;

    for (int tt = 0; tt < T_STEPS; ++tt) {
        const int t = dir ? (T_STEPS - 1 - tt) : tt;

        // ---- stage x_t into LDS (coalesced) + prefetch next step ----------
#pragma unroll
        for (int j = 0; j < 4; ++j) {
            Alds[rr[j] * ASTRIDE + uu[j]] =
                in[((size_t)(b0 + rr[j]) * T_STEPS + t) * UNITS_ + uu[j]];
        }
        if (tt + 1 < T_STEPS) {
            const int tn = dir ? (t - 1) : (t + 1);
            __builtin_prefetch(
                &in[((size_t)(b0 + rr[0]) * T_STEPS + tn) * UNITS_ + uu[0]], 0, 1);
        }
        __syncthreads();   // x_t and h_{t-1} visible to all waves

        // ---- z = [x_t ; h] @ [W;R]  via fp32 WMMA (K=4 per step) ----------
        v8f acc0 = {0.f, 0.f, 0.f, 0.f, 0.f, 0.f, 0.f, 0.f};
        v8f acc1 = {0.f, 0.f, 0.f, 0.f, 0.f, 0.f, 0.f, 0.f};
#pragma unroll 4
        for (int ks = 0; ks < KTOT / 4; ++ks) {
            v2f a  = *(const v2f*)(Alds + aBase + 4 * ks);
            v2f w0 = *(const v2f*)(pwt0 + ks * 64);
            v2f w1 = *(const v2f*)(pwt1 + ks * 64);
            acc0 = __builtin_amdgcn_wmma_f32_16x16x4_f32(
                false, a, false, w0, (short)0, acc0, false, false);
            acc1 = __builtin_amdgcn_wmma_f32_16x16x4_f32(
                false, a, false, w1, (short)0, acc1, false, false);
        }
        {   // scatter C/D layout (VGPR e, lane) -> z[row][col] in LDS
            const int zr = 8 * halfsel;
#pragma unroll
            for (int e = 0; e < 8; ++e) {
                Zlds[(zr + e) * ZSTRIDE + nt0 * 16 + lrow] = acc0[e];
                Zlds[(zr + e) * ZSTRIDE + nt1 * 16 + lrow] = acc1[e];
            }
        }
        __syncthreads();   // z complete

        // ---- LSTM cell elementwise: gates i,f,g,o; update c, h ------------
#pragma unroll
        for (int j = 0; j < 4; ++j) {
            const int r = rr[j], u = uu[j];
            float zi = Zlds[r * ZSTRIDE + u]       + bI[j];
            float zf = Zlds[r * ZSTRIDE + 256 + u] + bF[j];
            float zg = Zlds[r * ZSTRIDE + 512 + u] + bG[j];
            float zo = Zlds[r * ZSTRIDE + 768 + u] + bO[j];
            float gi = sigmoid_f(zi);
            float gf = sigmoid_f(zf);
            float gg = tanh_f(zg);
            float go = sigmoid_f(zo);
            float c  = gf * cst[j] + gi * gg;
            cst[j]   = c;
            float h  = go * tanh_f(c);
            Alds[r * ASTRIDE + 256 + u] = h;              // next-step recurrence
            float ov = residual ? (h + Alds[r * ASTRIDE + u]) : h;
            out[((size_t)(b0 + r) * T_STEPS + t) * UNITS_ + u] = ov;
        }
        __syncthreads();   // h written before next x-stage / GEMM
    }
}

// d_out holds the bw chain; final: out = (fw + out) * 0.5
__global__ void combine_avg_kernel(const float4* __restrict__ fw,
                                   float4* __restrict__ out, int n4) {
    int i = blockIdx.x * blockDim.x + threadIdx.x;
    if (i < n4) {
        float4 a = fw[i];
        float4 b = out[i];
        out[i] = make_float4((a.x + b.x) * 0.5f, (a.y + b.y) * 0.5f,
                             (a.z + b.z) * 0.5f, (a.w + b.w) * 0.5f);
    }
}

extern "C" void kernel_launch(void* const* d_in, const int* in_sizes, int n_in,
                              void* d_out, int out_size, void* d_ws, size_t ws_size,
                              hipStream_t stream) {
    const float* x   = (const float*)d_in[0];
    const float* kfw = (const float*)d_in[1];  // (2,256,1024)
    const float* rfw = (const float*)d_in[2];  // (2,256,1024)
    const float* bfw = (const float*)d_in[3];  // (2,1024)
    const float* kbw = (const float*)d_in[4];
    const float* rbw = (const float*)d_in[5];
    const float* bbw = (const float*)d_in[6];
    float* out = (float*)d_out;                // 128*256*256 fp32

    // workspace layout: packed weights (8 MB) + fw chain buffer (32 MB)
    float* pk   = (float*)d_ws;                     // 4 * COMBO_F floats
    float* wsfw = pk + (size_t)4 * COMBO_F;         // 8,388,608 floats
    float* wsbw = out;                              // bw chain lives in d_out

    // 1) pack [W;R] into WMMA-B lane order (one-time, deterministic)
    pack_weights_kernel<<<4096, 256, 0, stream>>>(kfw, rfw, kbw, rbw, pk);

    const size_t shmem = (size_t)(16 * ASTRIDE + 16 * ZSTRIDE) * sizeof(float);
    (void)hipFuncSetAttribute((const void*)bilstm_layer_kernel,
                              hipFuncAttributeMaxDynamicSharedMemorySize,
                              (int)shmem);

    // 2) layer 0 (both directions read x)
    bilstm_layer_kernel<<<dim3(8, 2), 1024, shmem, stream>>>(
        x, x, pk + 0 * (size_t)COMBO_F, pk + 1 * (size_t)COMBO_F,
        bfw + 0, bbw + 0, wsfw, wsbw, /*residual=*/0);

    // 3) layer 1, residual, safely in-place (each (b,t) read before written
    //    by the same workgroup at the same timestep)
    bilstm_layer_kernel<<<dim3(8, 2), 1024, shmem, stream>>>(
        wsfw, wsbw, pk + 2 * (size_t)COMBO_F, pk + 3 * (size_t)COMBO_F,
        bfw + 1024, bbw + 1024, wsfw, wsbw, /*residual=*/1);

    // 4) merge 'ave': out = (fw + bw) * 0.5   (bw already in d_out)
    const int n4 = (128 * 256 * 256) / 4;
    combine_avg_kernel<<<(n4 + 255) / 256, 256, 0, stream>>>(
        (const float4*)wsfw, (float4*)out, n4);
}